// Attention_12489764897521
// MI455X (gfx1250) — compile-verified
//
#include <hip/hip_runtime.h>
#include <hip/hip_bf16.h>

typedef __attribute__((ext_vector_type(2))) float v2f;
typedef __attribute__((ext_vector_type(8))) float v8f;

#define BATCH 32
#define HDIM 256
#define NDIM 10000
#define TILES_PER_B (NDIM / 16)      // 625 tiles of 16 columns
#define WAVES_PER_BLOCK 8            // 256 threads = 8 wave32
#define TOTAL_TASKS (BATCH * TILES_PER_B)        // 20000
#define NBLOCKS (TOTAL_TASKS / WAVES_PER_BLOCK)  // 2500 (exact)

__device__ __forceinline__ float fast_tanh(float x) {
#if __has_builtin(__builtin_amdgcn_tanhf)
    return __builtin_amdgcn_tanhf(x);   // v_tanh_f32 (gfx1250 TRANS op)
#else
    return tanhf(x);
#endif
}

__global__ __launch_bounds__(256) void Attention_12489764897521_kernel(
    const float* __restrict__ s,    // [B,H,N] static_hidden
    const float* __restrict__ d,    // [B,H,N] dynamic_hidden
    const float* __restrict__ dec,  // [B,H]   decoder_hidden
    const float* __restrict__ W,    // [1,H]
    float* __restrict__ out)        // [B,N]
{
    const int lane = threadIdx.x & 31;
    const int wave = threadIdx.x >> 5;
    const int task = blockIdx.x * WAVES_PER_BLOCK + wave;   // < 20000 exactly
    const int b    = task / TILES_PER_B;
    const int tile = task % TILES_PER_B;
    const int n0   = tile * 16;

    const int nl   = lane & 15;            // column within the 16-wide tile
    const int n    = n0 + nl;
    // Per the ISA 32-bit A/B layouts: lanes 0-15 carry K={0,1} in VGPR{0,1},
    // lanes 16-31 carry K={2,3}.
    const int k0   = (lane >> 4) * 2;      // 0 or 2
    const int k1   = k0 + 1;               // 1 or 3

    const float* __restrict__ sb   = s   + (size_t)b * HDIM * NDIM;
    const float* __restrict__ db   = d   + (size_t)b * HDIM * NDIM;
    const float* __restrict__ decb = dec + (size_t)b * HDIM;

    v8f c = {};   // f32 16x16 accumulator tile (8 VGPRs)

    #pragma unroll 4
    for (int h0 = 0; h0 < HDIM; h0 += 4) {
        const int ha = h0 + k0;
        const int hb = h0 + k1;
        const size_t ra = (size_t)ha * NDIM + n;
        const size_t rb = (size_t)hb * NDIM + n;

        // Single-use 655MB stream >> 192MB L2: use non-temporal loads.
        float s0 = __builtin_nontemporal_load(sb + ra);
        float d0 = __builtin_nontemporal_load(db + ra);
        float s1 = __builtin_nontemporal_load(sb + rb);
        float d1 = __builtin_nontemporal_load(db + rb);

        // B-matrix (4x16): tanh(dec + static + dynamic) for this lane's (h,n)
        v2f Bm;
        Bm.x = fast_tanh(decb[ha] + s0 + d0);
        Bm.y = fast_tanh(decb[hb] + s1 + d1);

        // A-matrix (16x4): W chunk replicated across all 16 rows -> every
        // D row accumulates the identical dot product over h.
        v2f Am;
        Am.x = W[ha];
        Am.y = W[hb];

        // D = A x B + C  (v_wmma_f32_16x16x4_f32), EXEC all-1s guaranteed.
        c = __builtin_amdgcn_wmma_f32_16x16x4_f32(
                /*neg_a=*/false, Am, /*neg_b=*/false, Bm,
                /*c_mod=*/(short)0, c, /*reuse_a=*/false, /*reuse_b=*/false);
    }

    // Row M=0 of D lives in VGPR0 (c[0]) of lanes 0-15, column = lane.
    if (lane < 16) {
        out[(size_t)b * NDIM + n] = c[0];
    }
}

extern "C" void kernel_launch(void* const* d_in, const int* in_sizes, int n_in,
                              void* d_out, int out_size, void* d_ws, size_t ws_size,
                              hipStream_t stream) {
    const float* s   = (const float*)d_in[0];   // static_hidden  [B,H,N]
    const float* d   = (const float*)d_in[1];   // dynamic_hidden [B,H,N]
    const float* dec = (const float*)d_in[2];   // decoder_hidden [B,H]
    const float* W   = (const float*)d_in[3];   // W [1,H]
    float* out = (float*)d_out;                 // [B,N]

    Attention_12489764897521_kernel<<<NBLOCKS, 256, 0, stream>>>(s, d, dec, W, out);
}